// EviPassingLayer_90494960926822
// MI455X (gfx1250) — compile-verified
//
#include <hip/hip_runtime.h>

// out[v] = sum_{e=(u->v)} h[u] * w[e]   (gather-multiply-scatter-add)
// h: [N,32] f32, w: [E] f32, src/dst: [E] int64, out: [N,32] f32.
//
// Wave32 mapping: lane == feature (D=32). One wave per contiguous chunk of
// 8 edges: merged scalar loads for edge records (one kmcnt wait per chunk),
// 8 independent coalesced 128B row gathers in flight, then 8 native f32
// atomics (device scope, no return) into the L2-resident output.

#define THREADS 256
#define WPB (THREADS / 32)   // waves per block
#define CHUNK 8              // edges per wave

__global__ __launch_bounds__(THREADS) void edge_scatter_kernel(
        const float* __restrict__ h,
        const float* __restrict__ w,
        const long long* __restrict__ src,
        const long long* __restrict__ dst,
        float* __restrict__ out,
        int n_edges) {
    const int lane = threadIdx.x & 31;
    // readfirstlane -> provably wave-uniform => scalar (SMEM) edge loads.
    const int wave = __builtin_amdgcn_readfirstlane(
        (int)(blockIdx.x * WPB + (threadIdx.x >> 5)));
    const int nChunks = n_edges >> 3;

    if (wave < nChunks) {
        const int base = wave << 3;

        // Phase 1: edge records. Adjacent uniform loads merge into wide
        // s_load (b512 for the int64 index runs, b256 for w); a single
        // s_wait_kmcnt covers the whole chunk.
        int   sv[CHUNK];
        int   dv[CHUNK];
        float wv[CHUNK];
#pragma unroll
        for (int i = 0; i < CHUNK; ++i) sv[i] = (int)src[base + i];
#pragma unroll
        for (int i = 0; i < CHUNK; ++i) dv[i] = (int)dst[base + i];
#pragma unroll
        for (int i = 0; i < CHUNK; ++i) wv[i] = w[base + i];

        // Phase 2: 8 independent coalesced row gathers (128B per wave each),
        // all outstanding on LOADcnt before any use.
        float hv[CHUNK];
#pragma unroll
        for (int i = 0; i < CHUNK; ++i)
            hv[i] = h[((size_t)sv[i] << 5) + lane];

        // Phase 3: scale + native f32 scatter-add (no return -> STOREcnt,
        // fire-and-forget into the L2 atomic units).
#pragma unroll
        for (int i = 0; i < CHUNK; ++i)
            __hip_atomic_fetch_add(&out[((size_t)dv[i] << 5) + lane],
                                   hv[i] * wv[i],
                                   __ATOMIC_RELAXED, __HIP_MEMORY_SCOPE_AGENT);
    } else {
        // Tail: waves beyond nChunks each take one leftover edge.
        const int e = nChunks * CHUNK + (wave - nChunks);
        if (e < n_edges) {
            const int   s  = (int)src[e];
            const int   d  = (int)dst[e];
            const float m  = h[((size_t)s << 5) + lane] * w[e];
            __hip_atomic_fetch_add(&out[((size_t)d << 5) + lane], m,
                                   __ATOMIC_RELAXED, __HIP_MEMORY_SCOPE_AGENT);
        }
    }
}

// Zero-fill with 16B stores (out_size is a multiple of 4 here; scalar tail
// guard kept for safety).
__global__ __launch_bounds__(THREADS) void zero_out_kernel(float* __restrict__ out, int n) {
    const int i4 = (blockIdx.x * blockDim.x + threadIdx.x) * 4;
    if (i4 + 3 < n) {
        *(float4*)(out + i4) = make_float4(0.f, 0.f, 0.f, 0.f);
    } else {
        for (int i = i4; i < n; ++i) out[i] = 0.0f;
    }
}

extern "C" void kernel_launch(void* const* d_in, const int* in_sizes, int n_in,
                              void* d_out, int out_size, void* d_ws, size_t ws_size,
                              hipStream_t stream) {
    const float*     h   = (const float*)d_in[0];       // [N*32]
    const float*     w   = (const float*)d_in[1];       // [E]
    const long long* src = (const long long*)d_in[2];   // [E] int64
    const long long* dst = (const long long*)d_in[3];   // [E] int64
    float*           out = (float*)d_out;               // [N*32]

    const int n_edges = in_sizes[2];
    const int n_out   = out_size;

    // Harness poisons d_out; accumulation needs zeros every call.
    const int zthreads = (n_out + 3) / 4;
    zero_out_kernel<<<(zthreads + THREADS - 1) / THREADS, THREADS, 0, stream>>>(out, n_out);

    // One wave per 8-edge chunk, plus one wave per tail edge.
    const int nChunks = n_edges / CHUNK;
    const int nWaves  = nChunks + (n_edges - nChunks * CHUNK);
    int blocks = (nWaves + WPB - 1) / WPB;
    if (blocks < 1) blocks = 1;
    edge_scatter_kernel<<<blocks, THREADS, 0, stream>>>(h, w, src, dst, out, n_edges);
}